// YoloLoss_80384607912711
// MI455X (gfx1250) — compile-verified
//
#include <hip/hip_runtime.h>
#include <math.h>

// ---------------- problem constants (match reference) ----------------
#define NUM_CLASSES 80
#define F_SIZE      26
#define F2          (F_SIZE * F_SIZE)   // 676
#define A_N         5
#define CELLS       (F2 * A_N)          // 3380 cell*anchor slots
#define T_N         32
#define OBJECT_SCALE 5.0f
#define BLOCK       256                 // 8 wave32s per workgroup

typedef __attribute__((ext_vector_type(2))) float v2f;
typedef __attribute__((ext_vector_type(8))) float v8f;

typedef __attribute__((address_space(1))) int* gint_p;   // global (device) int*
typedef __attribute__((address_space(3))) int* lint_p;   // LDS int*

#if __has_builtin(__builtin_amdgcn_wmma_f32_16x16x4_f32)
#define HAVE_WMMA_F32X4 1
#endif
#if __has_builtin(__builtin_amdgcn_global_load_async_to_lds_b32)
#define HAVE_ASYNC_LDS 1
#endif

// Reduce 32 per-lane f32 partials to their total (broadcast to all lanes).
// Uses V_WMMA_F32_16X16X4_F32: A = [s | 0] (s in K=0 cols for lanes 0-15,
// K=2 cols for lanes 16-31 per the 16x4 A layout), B = ones(4x16).
// => D[m][n] = s[m] + s[m+16] for every n. Each lane holds 8 D rows
// (M = 0..7 for lanes 0-15, M = 8..15 for lanes 16-31), so
// sum(d[0..7]) + shfl_xor(.,16) = exact f32 total.
__device__ __forceinline__ float wave32_total(float s) {
#ifdef HAVE_WMMA_F32X4
  v2f a; a[0] = s;    a[1] = 0.0f;
  v2f b; b[0] = 1.0f; b[1] = 1.0f;
  v8f c = {};
  v8f d = __builtin_amdgcn_wmma_f32_16x16x4_f32(
      /*neg_a=*/false, a, /*neg_b=*/false, b,
      /*c_mod=*/(short)0, c, /*reuse_a=*/false, /*reuse_b=*/false);
  float t = ((d[0] + d[1]) + (d[2] + d[3])) + ((d[4] + d[5]) + (d[6] + d[7]));
  t += __shfl_xor(t, 16);
  return t;
#else
  for (int off = 16; off > 0; off >>= 1) s += __shfl_xor(s, off);
  return s;
#endif
}

__global__ __launch_bounds__(BLOCK)
void yolo_loss_main(const float* __restrict__ feat,
                    const float* __restrict__ box_pred,
                    const float* __restrict__ box_conf,
                    const float* __restrict__ box_prob,
                    const float* __restrict__ targets,
                    const float* __restrict__ anchors,
                    float* __restrict__ partial) {
  __shared__ float  s_tgt[T_N * 5];      // raw targets (x1,y1,x2,y2,cls)
  __shared__ float  s_anc[A_N * 2];
  __shared__ float4 s_tb[T_N];           // target corner boxes
  __shared__ float  s_tarea[T_N];        // target areas (normalized coords)
  __shared__ float  s_m[T_N][5];         // scatter values (mvals)
  __shared__ int    s_posidx[T_N];       // flat cell*A_N + best_anchor
  __shared__ int    s_tab[CELLS];        // scatter table: target idx or -1
  __shared__ float  s_red[BLOCK];

  const int b   = (int)blockIdx.x;
  const int tid = (int)threadIdx.x;
  const float* tgt_g = targets + (size_t)b * (T_N * 5);

  // ---- stage targets into LDS (async global->LDS on gfx1250) ----
#ifdef HAVE_ASYNC_LDS
  if (tid < T_N * 5) {
    __builtin_amdgcn_global_load_async_to_lds_b32(
        (gint_p)(tgt_g + tid), (lint_p)&s_tgt[tid], 0, 0);
  }
#else
  if (tid < T_N * 5) s_tgt[tid] = tgt_g[tid];
#endif
  if (tid < A_N * 2) s_anc[tid] = anchors[tid];
  for (int i = tid; i < CELLS; i += BLOCK) s_tab[i] = -1;
#ifdef HAVE_ASYNC_LDS
#if __has_builtin(__builtin_amdgcn_s_wait_asynccnt)
  __builtin_amdgcn_s_wait_asynccnt(0);
#else
  asm volatile("s_wait_asynccnt 0" ::: "memory");
#endif
#endif
  __syncthreads();

  // ---- per-target derived values (one lane per target) ----
  if (tid < T_N) {
    const int t = tid;
    const float x1 = s_tgt[t * 5 + 0], y1 = s_tgt[t * 5 + 1];
    const float x2 = s_tgt[t * 5 + 2], y2 = s_tgt[t * 5 + 3];
    const float cls = s_tgt[t * 5 + 4];
    s_tb[t] = make_float4(x1, y1, x2, y2);
    s_tarea[t] = (x2 - x1) * (y2 - y1);
    const float tcx = (x1 + x2) * (0.5f * (float)F_SIZE);
    const float tcy = (y1 + y2) * (0.5f * (float)F_SIZE);
    const float tw  = (x2 - x1) * (float)F_SIZE;
    const float th  = (y2 - y1) * (float)F_SIZE;
    const float fx = floorf(tcx), fy = floorf(tcy);
    const int pos = (int)fy * F_SIZE + (int)fx;
    // best anchor: argmax IoU of origin-centered boxes (first max wins)
    int best = 0; float bestiou = -1.0f;
    const float ta = tw * th;
    #pragma unroll
    for (int a = 0; a < A_N; ++a) {
      const float aw = s_anc[a * 2 + 0], ah = s_anc[a * 2 + 1];
      const float inter = fminf(tw, aw) * fminf(th, ah);
      const float iou = inter / (ta + aw * ah - inter);
      if (iou > bestiou) { bestiou = iou; best = a; }
    }
    s_m[t][0] = tcx - fx;
    s_m[t][1] = tcy - fy;
    s_m[t][2] = logf(tw / s_anc[best * 2 + 0]);
    s_m[t][3] = logf(th / s_anc[best * 2 + 1]);
    s_m[t][4] = cls;
    s_posidx[t] = pos * A_N + best;
  }
  __syncthreads();
  // serial scatter: last write wins, matching sequential .at[].set order
  if (tid == 0) {
    for (int t = 0; t < T_N; ++t) s_tab[s_posidx[t]] = t;
  }
  __syncthreads();

  // ---- dense sweep over 3380 cell*anchor slots ----
  const float4* bp4 = (const float4*)box_pred + (size_t)b * CELLS;
  const float4* ft4 = (const float4*)feat     + (size_t)b * CELLS;
  const float*  cf  = box_conf + (size_t)b * CELLS;
  const float*  prb = box_prob + (size_t)b * CELLS * NUM_CLASSES;

  float acc_sq  = 0.0f;  // (conf + loc) squared sums, halved at the end
  float acc_cls = 0.0f;

  for (int idx = tid; idx < CELLS; idx += BLOCK) {
    const float4 p = bp4[idx];                       // (cx, cy, w, h)
    const float px1 = p.x - 0.5f * p.z, py1 = p.y - 0.5f * p.w;
    const float px2 = p.x + 0.5f * p.z, py2 = p.y + 0.5f * p.w;
    const float parea = p.z * p.w;
    float best = 0.0f;                               // IoU >= 0 always
    #pragma unroll 8
    for (int t = 0; t < T_N; ++t) {
      const float4 tb = s_tb[t];
      const float lx = fmaxf(px1, tb.x);
      const float ly = fmaxf(py1, tb.y);
      const float rx = fminf(px2, tb.z);
      const float ry = fminf(py2, tb.w);
      const float w = fmaxf(rx - lx, 0.0f), h = fmaxf(ry - ly, 0.0f);
      const float inter = w * h;
      best = fmaxf(best, inter / (parea + s_tarea[t] - inter));
    }
    const int   tsel = s_tab[idx];
    const float conf = cf[idx];
    float cmask, pmask;
    if (tsel >= 0) { cmask = OBJECT_SCALE; pmask = 1.0f; }
    else           { cmask = (best > 0.5f) ? 0.0f : 1.0f; pmask = 0.0f; }
    const float cterm = cmask * (conf - pmask);
    acc_sq += cterm * cterm;

    if (tsel >= 0) {                                 // <=32 lanes per batch
      const float4 f = ft4[idx];
      const float bx = 1.0f / (1.0f + expf(-f.x));
      const float by = 1.0f / (1.0f + expf(-f.y));
      const float d0 = bx  - s_m[tsel][0];
      const float d1 = by  - s_m[tsel][1];
      const float d2 = f.z - s_m[tsel][2];
      const float d3 = f.w - s_m[tsel][3];
      acc_sq += d0 * d0 + d1 * d1 + d2 * d2 + d3 * d3;
      // class loss: logsumexp over 80 classes, only at scattered positions
      const float* pr = prb + (size_t)idx * NUM_CLASSES;
      const int label = (int)s_m[tsel][4];
      float mx = pr[0];
      for (int c2 = 1; c2 < NUM_CLASSES; ++c2) mx = fmaxf(mx, pr[c2]);
      float se = 0.0f;
      for (int c2 = 0; c2 < NUM_CLASSES; ++c2) se += expf(pr[c2] - mx);
      acc_cls += (mx + logf(se)) - pr[label];
    }
  }

  // ---- deterministic block reduction (LDS fold + WMMA wave reduce) ----
  s_red[tid] = 0.5f * acc_sq + acc_cls;
  __syncthreads();
  if (tid < 32) {
    float s = 0.0f;
    #pragma unroll
    for (int i = 0; i < BLOCK / 32; ++i) s += s_red[tid + i * 32];
    const float tot = wave32_total(s);   // EXEC all-ones within wave 0
    if (tid == 0) partial[b] = tot;
  }
}

__global__ __launch_bounds__(128)
void yolo_loss_final(const float* __restrict__ partial,
                     float* __restrict__ out, int nb) {
  __shared__ float s_red[128];
  const int tid = (int)threadIdx.x;
  s_red[tid] = (tid < nb) ? partial[tid] : 0.0f;
  __syncthreads();
  if (tid < 32) {
    float s = s_red[tid] + s_red[tid + 32] + s_red[tid + 64] + s_red[tid + 96];
    const float tot = wave32_total(s);
    if (tid == 0) out[0] = tot / (float)nb;
  }
}

extern "C" void kernel_launch(void* const* d_in, const int* in_sizes, int n_in,
                              void* d_out, int out_size, void* d_ws, size_t ws_size,
                              hipStream_t stream) {
  const float* feat     = (const float*)d_in[0];
  const float* box_pred = (const float*)d_in[1];
  const float* box_conf = (const float*)d_in[2];
  const float* box_prob = (const float*)d_in[3];
  const float* targets  = (const float*)d_in[4];
  const float* anchors  = (const float*)d_in[5];
  const int B = in_sizes[0] / (CELLS * 4);   // 128

  float* partial = (float*)d_ws;             // B floats of scratch
  yolo_loss_main<<<B, BLOCK, 0, stream>>>(feat, box_pred, box_conf, box_prob,
                                          targets, anchors, partial);
  yolo_loss_final<<<1, 128, 0, stream>>>(partial, (float*)d_out, B);
}